// VQLinearEmbedding_42133629174448
// MI455X (gfx1250) — compile-verified
//
#include <hip/hip_runtime.h>

typedef __attribute__((ext_vector_type(2))) float v2f;
typedef __attribute__((ext_vector_type(4))) float v4f;
typedef __attribute__((ext_vector_type(8))) float v8f;

#define KCODES 512
#define DDIM 64
#define LDS_PITCH 68   // 68 floats/row: 16B row alignment + bank-conflict-free B loads

__global__ __launch_bounds__(256) void vq_argmin_wmma(const float* __restrict__ z,
                                                      const float* __restrict__ cb,
                                                      int* __restrict__ out,
                                                      int n_rows) {
  extern __shared__ float smem[];               // [512*68] codebook + [512] norms
  float* cbl   = smem;
  float* cnorm = smem + KCODES * LDS_PITCH;

  const int tid  = threadIdx.x;
  const int lane = tid & 31;
  const int wave = tid >> 5;

  // ---- stage codebook HBM -> LDS (float4 chunks) ----
  for (int idx = tid; idx < KCODES * (DDIM / 4); idx += 256) {
    int row = idx >> 4;                  // 16 float4 chunks per 64-float row
    int c4  = (idx & 15) << 2;
    v4f v = *(const v4f*)(cb + row * DDIM + c4);
    *(v4f*)(cbl + row * LDS_PITCH + c4) = v;
  }
  __syncthreads();

  // ---- codebook squared norms ----
  for (int row = tid; row < KCODES; row += 256) {
    const float* r = cbl + row * LDS_PITCH;
    float s = 0.f;
#pragma unroll
    for (int c = 0; c < DDIM; c += 4) {
      v4f v = *(const v4f*)(r + c);
      s += v.x * v.x + v.y * v.y + v.z * v.z + v.w * v.w;
    }
    cnorm[row] = s;
  }
  __syncthreads();

  // ---- per-wave 16-row tile of z ----
  const int rowBase = blockIdx.x * 128 + wave * 16;
  const int m = lane & 15;
  const int h = lane >> 4;      // ISA A-layout: half 0 -> K pairs {0,1}, half 1 -> {2,3}
  int arow = rowBase + m;
  if (arow >= n_rows) arow = n_rows - 1;   // clamp: keep EXEC all-1s for WMMA

  v2f afrag[16];
#pragma unroll
  for (int kk = 0; kk < 16; ++kk)
    afrag[kk] = *(const v2f*)(z + (size_t)arow * DDIM + kk * 4 + h * 2);

  float bestv[8];
  int   besti[8];
#pragma unroll
  for (int r = 0; r < 8; ++r) { bestv[r] = 3.4e38f; besti[r] = 0; }

  // Two 16-column code tiles per iteration: independent accumulator chains,
  // shared A fragments, halved loop overhead per WMMA.
  for (int ct = 0; ct < KCODES / 32; ++ct) {
    const float* b0 = cbl + (ct * 32 + m) * LDS_PITCH + h * 2;   // B = codebook^T tiles
    const float* b1 = b0 + 16 * LDS_PITCH;
    v8f acc0 = {};
    v8f acc1 = {};
#pragma unroll
    for (int kk = 0; kk < 16; ++kk) {
      v2f bb0 = *(const v2f*)(b0 + kk * 4);
      v2f bb1 = *(const v2f*)(b1 + kk * 4);
      // 8 args: (neg_a, A, neg_b, B, c_mod, C, reuse_a, reuse_b)
      acc0 = __builtin_amdgcn_wmma_f32_16x16x4_f32(false, afrag[kk], false, bb0,
                                                   (short)0, acc0, false, false);
      acc1 = __builtin_amdgcn_wmma_f32_16x16x4_f32(false, afrag[kk], false, bb1,
                                                   (short)0, acc1, false, false);
    }
    const int   code0 = ct * 32 + m;
    const int   code1 = code0 + 16;
    const float cn0   = cnorm[code0];
    const float cn1   = cnorm[code1];
#pragma unroll
    for (int r = 0; r < 8; ++r) {
      // ||e||^2 - 2 z.e  (||z||^2 constant per row, irrelevant to argmin).
      // Candidate code index is strictly increasing within a lane, so a
      // strict < keeps the first-occurrence minimum — no tie-break needed here.
      float d0 = fmaf(-2.0f, acc0[r], cn0);
      if (d0 < bestv[r]) { bestv[r] = d0; besti[r] = code0; }
      float d1 = fmaf(-2.0f, acc1[r], cn1);
      if (d1 < bestv[r]) { bestv[r] = d1; besti[r] = code1; }
    }
  }

  // ---- reduce argmin across the 16 lanes of the N dimension ----
  // (tie-break on smaller index to match jnp.argmin first-occurrence rule)
#pragma unroll
  for (int off = 8; off >= 1; off >>= 1) {
#pragma unroll
    for (int r = 0; r < 8; ++r) {
      float ov = __shfl_xor(bestv[r], off, 32);
      int   oi = __shfl_xor(besti[r], off, 32);
      if (ov < bestv[r] || (ov == bestv[r] && oi < besti[r])) {
        bestv[r] = ov; besti[r] = oi;
      }
    }
  }

  if (m == 0) {
#pragma unroll
    for (int r = 0; r < 8; ++r) {
      int row = rowBase + h * 8 + r;     // lanes 0/16 hold M = r / 8+r
      if (row < n_rows) out[row] = besti[r];
    }
  }
}

extern "C" void kernel_launch(void* const* d_in, const int* in_sizes, int n_in,
                              void* d_out, int out_size, void* d_ws, size_t ws_size,
                              hipStream_t stream) {
  const float* z  = (const float*)d_in[0];
  const float* cb = (const float*)d_in[1];
  int* out = (int*)d_out;

  int n_rows = in_sizes[0] / DDIM;                 // 32*4096 = 131072
  int blocks = (n_rows + 127) / 128;               // 128 rows per block (8 waves x 16)
  size_t shmem = (size_t)(KCODES * LDS_PITCH + KCODES) * sizeof(float);  // ~138 KB < 320 KB

  hipLaunchKernelGGL(vq_argmin_wmma, dim3(blocks), dim3(256), shmem, stream,
                     z, cb, out, n_rows);
}